// noattentionplus_77687368450112
// MI455X (gfx1250) — compile-verified
//
#include <hip/hip_runtime.h>

typedef __attribute__((ext_vector_type(2))) float v2f;
typedef __attribute__((ext_vector_type(8))) float v8f;

// Workspace layout (floats):
//   [0 .. 3071]  Wm packed into V_WMMA_F32_16X16X4_F32 B-operand lane layout
//                (2 N-tiles x 24 K-chunks x 32 lanes x 2 floats)
//   [3072..3161] Weff (5x18)  = W3 @ W2 @ W1   (genre MLP collapsed: no activations)
//   [3162..3166] beff (5)     = W3 @ (W2 @ b1 + b2) + b3
#define WS_PACKED 0
#define WS_WEFF   3072
#define WS_BEFF   (3072 + 90)

// ---------------------------------------------------------------------------
// Prep kernel: collapse the activation-free genre MLP and pre-swizzle Wm into
// the WMMA B-operand register layout. Runs as a single 256-thread block.
// ---------------------------------------------------------------------------
__global__ __launch_bounds__(256) void noatt_prep_kernel(
    const float* __restrict__ Wm,   // [32][96]
    const float* __restrict__ W1, const float* __restrict__ b1,   // [30][18],[30]
    const float* __restrict__ W2, const float* __restrict__ b2,   // [30][30],[30]
    const float* __restrict__ W3, const float* __restrict__ b3,   // [5][30],[5]
    float* __restrict__ ws)
{
    __shared__ float sW21[30 * 18];
    __shared__ float sbt[30];
    const int tid = threadIdx.x;

    for (int idx = tid; idx < 540; idx += 256) {
        int r = idx / 18, c = idx % 18;
        float acc = 0.f;
        for (int k = 0; k < 30; ++k) acc += W2[r * 30 + k] * W1[k * 18 + c];
        sW21[idx] = acc;
    }
    for (int idx = tid; idx < 30; idx += 256) {
        float acc = b2[idx];
        for (int k = 0; k < 30; ++k) acc += W2[idx * 30 + k] * b1[k];
        sbt[idx] = acc;
    }
    // Pack Wm^T (96x32) into B-operand layout: for chunk kc, N-tile n,
    // lane L, vgpr v holds B[k = 4*kc + 2*(L/16) + v][N = 16*n + L%16],
    // where B[k][N] = Wm[N][k].
    for (int idx = tid; idx < 3072; idx += 256) {
        int n    = idx / 1536;
        int rem  = idx % 1536;
        int kc   = rem / 64;
        int t    = rem % 64;
        int lane = t >> 1;
        int v    = t & 1;
        int K = 4 * kc + 2 * (lane >> 4) + v;
        int N = 16 * n + (lane & 15);
        ws[WS_PACKED + idx] = Wm[N * 96 + K];
    }
    __syncthreads();
    for (int idx = tid; idx < 90; idx += 256) {
        int r = idx / 18, c = idx % 18;
        float acc = 0.f;
        for (int k = 0; k < 30; ++k) acc += W3[r * 30 + k] * sW21[k * 18 + c];
        ws[WS_WEFF + idx] = acc;
    }
    for (int idx = tid; idx < 5; idx += 256) {
        float acc = b3[idx];
        for (int k = 0; k < 30; ++k) acc += W3[idx * 30 + k] * sbt[k];
        ws[WS_BEFF + idx] = acc;
    }
}

// ---------------------------------------------------------------------------
// Main kernel: 256 threads = 8 waves; each wave32 owns 32 queries.
// grp_e = mem_e(96) @ Wm^T via 96x v_wmma_f32_16x16x4_f32 per wave-tile
// (24 K-chunks x 2 M-tiles x 2 N-tiles), fp32 end to end.
// ---------------------------------------------------------------------------
__global__ __launch_bounds__(256) void noatt_main_kernel(
    const int*   __restrict__ group_inputs,
    const int*   __restrict__ item_inputs,
    const int*   __restrict__ group_members, // [NUM_GROUPS][3]
    const float* __restrict__ user_emb,      // [NUM_USERS][32]
    const float* __restrict__ item_emb,      // [NUM_ITEMS][27]
    const float* __restrict__ genres,        // [NUM_ITEMS][18]
    const float* __restrict__ bm,            // [32]
    const float* __restrict__ Wp1,           // [8][96]
    const float* __restrict__ bp1,           // [8]
    const float* __restrict__ Wp2,           // [8]
    const float* __restrict__ bp2,           // [1]
    const float* __restrict__ ws,
    float* __restrict__ out, int Bq)
{
    __shared__ float sWp1[8 * 96];
    __shared__ float sWeff[5 * 18];
    __shared__ float sbeff[5];
    __shared__ float sbp1[8];
    __shared__ float sWp2[8];
    __shared__ float sbm[32];
    __shared__ float sGrp[8][32 * 33];   // per-wave transpose tile, pad-33 -> bank-conflict free

    const int tid = threadIdx.x;
    for (int i = tid; i < 768; i += 256) sWp1[i] = Wp1[i];
    for (int i = tid; i < 90;  i += 256) sWeff[i] = ws[WS_WEFF + i];
    if (tid < 5)  sbeff[tid] = ws[WS_BEFF + tid];
    if (tid < 8)  { sbp1[tid] = bp1[tid]; sWp2[tid] = Wp2[tid]; }
    if (tid < 32) sbm[tid] = bm[tid];
    __syncthreads();

    const int wave  = tid >> 5;
    const int lane  = tid & 31;
    const int lo    = lane & 15;
    const int g2    = (lane >> 4) << 1;          // 0 or 2: K sub-offset per A layout
    const int qbase = blockIdx.x * 256 + wave * 32;
    const int q     = min(qbase + lane, Bq - 1); // clamp: full EXEC through WMMA region

    // Prefetch the streaming index arrays well ahead (global_prefetch_b8).
    __builtin_prefetch(group_inputs + min(q + 16384, Bq - 1), 0, 0);
    __builtin_prefetch(item_inputs  + min(q + 16384, Bq - 1), 0, 0);

    const int grp = group_inputs[q];
    const int it  = item_inputs[q];
    const int m0 = group_members[grp * 3 + 0];
    const int m1 = group_members[grp * 3 + 1];
    const int m2 = group_members[grp * 3 + 2];

    // Broadcast member row ids: M-tile0 = queries qbase+0..15, M-tile1 = +16..31.
    const int uA0 = __shfl(m0, lo),      uA1 = __shfl(m1, lo),      uA2 = __shfl(m2, lo);
    const int uB0 = __shfl(m0, 16 + lo), uB1 = __shfl(m1, 16 + lo), uB2 = __shfl(m2, 16 + lo);

    const float* pA[3] = { user_emb + (size_t)uA0 * 32 + g2,
                           user_emb + (size_t)uA1 * 32 + g2,
                           user_emb + (size_t)uA2 * 32 + g2 };
    const float* pB[3] = { user_emb + (size_t)uB0 * 32 + g2,
                           user_emb + (size_t)uB1 * 32 + g2,
                           user_emb + (size_t)uB2 * 32 + g2 };
    const float* wp = ws + WS_PACKED + lane * 2;

    v8f acc0 = {}, acc1 = {}, acc2 = {}, acc3 = {};
    #pragma unroll
    for (int kc = 0; kc < 24; ++kc) {
        const int w = kc >> 3;                 // which member embedding (row of 32)
        const int c = (kc & 7) << 2;           // column within the 32-float row
        v2f a0 = *(const v2f*)(pA[w] + c);     // A, M-tile0 (16x4 fp32 operand)
        v2f a1 = *(const v2f*)(pB[w] + c);     // A, M-tile1
        v2f b0 = *(const v2f*)(wp + kc * 64);          // B, N-tile0 (dims 0..15)
        v2f b1 = *(const v2f*)(wp + (24 + kc) * 64);   // B, N-tile1 (dims 16..31)
        acc0 = __builtin_amdgcn_wmma_f32_16x16x4_f32(false, a0, false, b0, (short)0, acc0, false, false);
        acc1 = __builtin_amdgcn_wmma_f32_16x16x4_f32(false, a0, false, b1, (short)0, acc1, false, false);
        acc2 = __builtin_amdgcn_wmma_f32_16x16x4_f32(false, a1, false, b0, (short)0, acc2, false, false);
        acc3 = __builtin_amdgcn_wmma_f32_16x16x4_f32(false, a1, false, b1, (short)0, acc3, false, false);
    }

    // Add bias and transpose through LDS: C/D layout is (M = v + 8*(lane/16), N = lane%16).
    const float blo = sbm[lo], bhi = sbm[16 + lo];
    float* gdst = sGrp[wave];
    #pragma unroll
    for (int v = 0; v < 8; ++v) {
        const int m = v + ((lane >> 4) << 3);
        gdst[m * 33 + lo]             = acc0[v] + blo;
        gdst[m * 33 + 16 + lo]        = acc1[v] + bhi;
        gdst[(16 + m) * 33 + lo]      = acc2[v] + blo;
        gdst[(16 + m) * 33 + 16 + lo] = acc3[v] + bhi;
    }
    __syncthreads();

    // ---- per-lane epilogue: one query per lane ----
    // genre affine (collapsed MLP): ge = Weff @ g + beff
    float ge[5];
    {
        float gv[18];
        const float* gp = genres + (size_t)it * 18;
        #pragma unroll
        for (int j = 0; j < 18; ++j) gv[j] = gp[j];
        #pragma unroll
        for (int r = 0; r < 5; ++r) {
            float acc = sbeff[r];
            #pragma unroll
            for (int j = 0; j < 18; ++j) acc += sWeff[r * 18 + j] * gv[j];
            ge[r] = acc;
        }
    }
    float ie[32];
    {
        const float* ip = item_emb + (size_t)it * 27;
        #pragma unroll
        for (int d = 0; d < 27; ++d) ie[d] = ip[d];
        #pragma unroll
        for (int r = 0; r < 5; ++r) ie[27 + r] = ge[r];
    }
    float gr[32];
    {
        const float* gsrc = sGrp[wave] + lane * 33;
        #pragma unroll
        for (int d = 0; d < 32; ++d) gr[d] = gsrc[d];
    }

    float h[8];
    #pragma unroll
    for (int j = 0; j < 8; ++j) h[j] = sbp1[j];
    #pragma unroll
    for (int d = 0; d < 32; ++d) {
        const float e = gr[d] * ie[d];
        #pragma unroll
        for (int j = 0; j < 8; ++j)
            h[j] += sWp1[j * 96 + d] * e
                  + sWp1[j * 96 + 32 + d] * gr[d]
                  + sWp1[j * 96 + 64 + d] * ie[d];
    }
    float y = bp2[0];
    #pragma unroll
    for (int j = 0; j < 8; ++j) y += sWp2[j] * fmaxf(h[j], 0.f);
    y = 1.f / (1.f + __expf(-y));

    const int qout = qbase + lane;
    if (qout < Bq) out[qout] = y;
}

// ---------------------------------------------------------------------------
extern "C" void kernel_launch(void* const* d_in, const int* in_sizes, int n_in,
                              void* d_out, int out_size, void* d_ws, size_t ws_size,
                              hipStream_t stream)
{
    const int*   group_inputs  = (const int*)d_in[0];
    const int*   item_inputs   = (const int*)d_in[1];
    const int*   group_members = (const int*)d_in[2];
    const float* user_emb      = (const float*)d_in[3];
    const float* item_emb      = (const float*)d_in[4];
    const float* genres        = (const float*)d_in[5];
    const float* Wm            = (const float*)d_in[6];
    const float* bm            = (const float*)d_in[7];
    const float* W1            = (const float*)d_in[8];
    const float* b1            = (const float*)d_in[9];
    const float* W2            = (const float*)d_in[10];
    const float* b2            = (const float*)d_in[11];
    const float* W3            = (const float*)d_in[12];
    const float* b3            = (const float*)d_in[13];
    const float* Wp1           = (const float*)d_in[14];
    const float* bp1           = (const float*)d_in[15];
    const float* Wp2           = (const float*)d_in[16];
    const float* bp2           = (const float*)d_in[17];
    float* out = (float*)d_out;
    float* ws  = (float*)d_ws;
    const int Bq = in_sizes[0];

    noatt_prep_kernel<<<1, 256, 0, stream>>>(Wm, W1, b1, W2, b2, W3, b3, ws);

    const int nblocks = (Bq + 255) / 256;
    noatt_main_kernel<<<nblocks, 256, 0, stream>>>(
        group_inputs, item_inputs, group_members, user_emb, item_emb, genres,
        bm, Wp1, bp1, Wp2, bp2, ws, out, Bq);
}